// GraphAttention_29738353557518
// MI455X (gfx1250) — compile-verified
//
#include <hip/hip_runtime.h>
#include <hip/hip_bf16.h>
#include <math.h>

typedef float v2f __attribute__((ext_vector_type(2)));
typedef float v8f __attribute__((ext_vector_type(8)));

#define FEAT 32

// ---------------------------------------------------------------------------
// Phase 0: zero the output accumulator and the softmax denominators.
// ---------------------------------------------------------------------------
__global__ void gat_zero(float* __restrict__ out, int nOut,
                         float* __restrict__ denom, int nDen) {
  int i = blockIdx.x * blockDim.x + threadIdx.x;
  if (i < nOut) out[i] = 0.0f;
  if (i < nDen) denom[i] = 0.0f;
}

// ---------------------------------------------------------------------------
// Phase 1: H[rows x 32] = X[rows x 32] * W[32 x 32] using V_WMMA_F32_16X16X4_F32.
// One wave computes one 16-row M-tile against both 16-col N-tiles (16 WMMAs).
// rows = N_NODES * BATCH (features is [N, B, F] row-major == [rows, 32]).
// ---------------------------------------------------------------------------
__global__ void gat_gemm_wmma(const float* __restrict__ X,
                              const float* __restrict__ W,
                              float* __restrict__ H, int rows) {
  const int lane  = threadIdx.x & 31;
  const int wave  = threadIdx.x >> 5;
  const int tile  = blockIdx.x * (blockDim.x >> 5) + wave;
  const int nTiles = (rows + 15) >> 4;
  if (tile >= nTiles) return;          // wave-uniform: EXEC stays all-ones

  const int rlo   = lane & 15;         // M (for A/store) or N (for B)
  const int khalf = lane >> 4;         // which K-pair this lane holds

  int arow = tile * 16 + rlo;
  if (arow >= rows) arow = rows - 1;   // clamp; stores are guarded below

  v8f acc0 = {};                       // cols 0..15
  v8f acc1 = {};                       // cols 16..31

#pragma unroll
  for (int k0 = 0; k0 < FEAT; k0 += 4) {
    const int kb = k0 + khalf * 2;
    // A 16x4 f32: lanes 0-15 -> K=k0+0,1 ; lanes 16-31 -> K=k0+2,3
    v2f a = *(const v2f*)(X + (size_t)arow * FEAT + kb);
    // B 4x16 f32: lane gives N, VGPR0/1 give the lane's two K values
    v2f b0, b1;
    b0.x = W[(kb + 0) * FEAT + rlo];
    b0.y = W[(kb + 1) * FEAT + rlo];
    b1.x = W[(kb + 0) * FEAT + 16 + rlo];
    b1.y = W[(kb + 1) * FEAT + 16 + rlo];
    acc0 = __builtin_amdgcn_wmma_f32_16x16x4_f32(false, a, false, b0,
                                                 (short)0, acc0, false, false);
    acc1 = __builtin_amdgcn_wmma_f32_16x16x4_f32(false, a, false, b1,
                                                 (short)0, acc1, false, false);
  }

  // C/D 16x16 f32: VGPR r, lanes 0-15 -> M=r ; lanes 16-31 -> M=r+8
#pragma unroll
  for (int r = 0; r < 8; ++r) {
    const int m = tile * 16 + r + khalf * 8;
    if (m < rows) {
      H[(size_t)m * FEAT + rlo]      = acc0[r];
      H[(size_t)m * FEAT + 16 + rlo] = acc1[r];
    }
  }
}

// ---------------------------------------------------------------------------
// Phase 2: per-node attention scalars. s_src[r] = h[r]·a[:32], s_dst[r] = h[r]·a[32:].
// Linearity of the logit lets us collapse per-edge dot products to one add.
// ---------------------------------------------------------------------------
__global__ void gat_scores(const float* __restrict__ H,
                           const float* __restrict__ a,
                           float* __restrict__ s_src,
                           float* __restrict__ s_dst, int rows) {
  int i = blockIdx.x * blockDim.x + threadIdx.x;
  if (i >= rows) return;
  const float* hp = H + (size_t)i * FEAT;
  float ss = 0.0f, sd = 0.0f;
#pragma unroll
  for (int f = 0; f < FEAT; ++f) {
    float v = hp[f];
    ss += v * a[f];
    sd += v * a[FEAT + f];
  }
  s_src[i] = ss;
  s_dst[i] = sd;
}

// ---------------------------------------------------------------------------
// Phase 3: softmax denominators. One thread per (edge, batch).
// e = s_src[src] + s_dst[dst]; leaky-relu(0.2); denom[dst] += exp(e).
// ---------------------------------------------------------------------------
__global__ void gat_edge_denom(const int* __restrict__ src,
                               const int* __restrict__ dst,
                               const float* __restrict__ s_src,
                               const float* __restrict__ s_dst,
                               float* __restrict__ denom, int nEdges) {
  int i = blockIdx.x * blockDim.x + threadIdx.x;
  int e = i >> 1;
  int b = i & 1;
  if (e >= nEdges) return;
  int sRow = src[e] * 2 + b;
  int dRow = dst[e] * 2 + b;
  float ev = s_src[sRow] + s_dst[dRow];
  ev = ev > 0.0f ? ev : 0.2f * ev;
  atomicAdd(&denom[dRow], expf(ev));
}

// ---------------------------------------------------------------------------
// Phase 4: weighted scatter. One wave per edge, lane = feature.
// alpha recomputed from s arrays (wave-uniform broadcast loads, L2-resident).
// ---------------------------------------------------------------------------
__global__ void gat_edge_scatter(const int* __restrict__ src,
                                 const int* __restrict__ dst,
                                 const float* __restrict__ s_src,
                                 const float* __restrict__ s_dst,
                                 const float* __restrict__ H,
                                 const float* __restrict__ denom,
                                 float* __restrict__ out, int nEdges) {
  int i = blockIdx.x * blockDim.x + threadIdx.x;
  int e = i >> 5;
  int f = i & 31;
  if (e >= nEdges) return;
  int sn = src[e];
  int dn = dst[e];
#pragma unroll
  for (int b = 0; b < 2; ++b) {
    int sRow = sn * 2 + b;
    int dRow = dn * 2 + b;
    float ev = s_src[sRow] + s_dst[dRow];
    ev = ev > 0.0f ? ev : 0.2f * ev;
    float alpha = expf(ev) / denom[dRow];
    atomicAdd(&out[(size_t)dRow * FEAT + f], alpha * H[(size_t)sRow * FEAT + f]);
  }
}

// ---------------------------------------------------------------------------
// Phase 5: ELU epilogue in place.
// ---------------------------------------------------------------------------
__global__ void gat_elu(float* __restrict__ out, int n) {
  int i = blockIdx.x * blockDim.x + threadIdx.x;
  if (i >= n) return;
  float x = out[i];
  out[i] = x > 0.0f ? x : (expf(x) - 1.0f);
}

// ---------------------------------------------------------------------------
extern "C" void kernel_launch(void* const* d_in, const int* in_sizes, int n_in,
                              void* d_out, int out_size, void* d_ws, size_t ws_size,
                              hipStream_t stream) {
  (void)n_in; (void)ws_size;
  const float* features = (const float*)d_in[0];   // [N, B, 32]
  const float* W        = (const float*)d_in[1];   // [32, 32]
  const float* a        = (const float*)d_in[2];   // [64, 1]
  const int*   edge_src = (const int*)d_in[3];     // [E]
  const int*   edge_dst = (const int*)d_in[4];     // [E]

  const int nEdges = in_sizes[3];
  const int rows   = in_sizes[0] / FEAT;           // N_NODES * BATCH

  // Workspace layout (~14 MB): H | s_src | s_dst | denom
  float* H      = (float*)d_ws;
  float* s_src  = H + (size_t)rows * FEAT;
  float* s_dst  = s_src + rows;
  float* denom  = s_dst + rows;
  float* out    = (float*)d_out;

  // Phase 0: zero accumulators
  {
    int n = out_size > rows ? out_size : rows;
    gat_zero<<<(n + 255) / 256, 256, 0, stream>>>(out, out_size, denom, rows);
  }
  // Phase 1: H = X * W (WMMA f32 16x16x4)
  {
    int nTiles = (rows + 15) / 16;
    int blocks = (nTiles + 7) / 8;                 // 8 waves (256 thr) per block
    gat_gemm_wmma<<<blocks, 256, 0, stream>>>(features, W, H, rows);
  }
  // Phase 2: per-node attention scalars
  gat_scores<<<(rows + 255) / 256, 256, 0, stream>>>(H, a, s_src, s_dst, rows);
  // Phase 3: softmax denominators
  {
    int t = nEdges * 2;
    gat_edge_denom<<<(t + 255) / 256, 256, 0, stream>>>(edge_src, edge_dst,
                                                        s_src, s_dst, denom, nEdges);
  }
  // Phase 4: weighted scatter (wave per edge)
  {
    long long t = (long long)nEdges * 32;
    gat_edge_scatter<<<(int)((t + 255) / 256), 256, 0, stream>>>(
        edge_src, edge_dst, s_src, s_dst, H, denom, out, nEdges);
  }
  // Phase 5: ELU
  gat_elu<<<(out_size + 255) / 256, 256, 0, stream>>>(out, out_size);
}